// REN_72138270704170
// MI455X (gfx1250) — compile-verified
//
#include <hip/hip_runtime.h>
#include <math.h>

#define NBATCH 16384
#define NX 256
#define NY 128
#define NU 128
#define NQ 128

typedef __attribute__((ext_vector_type(2))) float v2f;
typedef __attribute__((ext_vector_type(8))) float v8f;
typedef int v4i_vs __attribute__((vector_size(16)));   // matches builtin pointee

struct AChunks { const float* p[8]; int ld[8]; };

// ---------------------------------------------------------------------------
// CDNA5 async global->LDS copy (ASYNCcnt path), with sync fallback so the
// file compiles on toolchains lacking the builtin.
// Builtin signature (from hipcc diagnostic): param0 = v4i_vs __device__*,
// i.e. (global src, lds dst, imm offset, imm cpol).
// ---------------------------------------------------------------------------
__device__ __forceinline__ void copy16_async(const float* __restrict__ g, float* l) {
#if __has_builtin(__builtin_amdgcn_global_load_async_to_lds_b128)
  __builtin_amdgcn_global_load_async_to_lds_b128(
      (__attribute__((address_space(1))) v4i_vs*)g,
      (__attribute__((address_space(3))) v4i_vs*)l, 0, 0);
#else
  *(float4*)l = *(const float4*)g;
#endif
}

__device__ __forceinline__ void wait_async0() {
#if __has_builtin(__builtin_amdgcn_s_wait_asynccnt)
  __builtin_amdgcn_s_wait_asynccnt(0);
#else
  asm volatile("s_wait_asynccnt 0x0" ::: "memory");
#endif
}

// ---------------------------------------------------------------------------
// Swizzled B layout for V_WMMA_F32_16X16X4_F32:
//   B (4x16) fragment: VGPR0 lanes0-15 = K=k0+0, lanes16-31 = K=k0+2
//                      VGPR1 lanes0-15 = K=k0+1, lanes16-31 = K=k0+3
//   Stored as v2f per lane so a fragment load is one coalesced global_load_b64.
// ---------------------------------------------------------------------------
__device__ __forceinline__ size_t bsw_idx(int k, int n, int nftot) {
  int kstep = k >> 2, kk = k & 3, nfrag = n >> 4, nr = n & 15;
  int lane = ((kk >> 1) << 4) | nr;   // kk in {2,3} -> upper half-wave
  int slot = kk & 1;                  // kk odd -> second float of the pair
  return (((size_t)kstep * nftot + nfrag) * 32 + lane) * 2 + slot;
}

// ---------------- prep kernels ----------------
__global__ __launch_bounds__(256) void init_X_kernel(float* __restrict__ X) {
  int i = blockIdx.x * 256 + threadIdx.x;          // 256*256 elements
  X[i] = ((i >> 8) == (i & 255)) ? 1.0f : 0.0f;    // X0 = I
}

__global__ __launch_bounds__(256) void fill_bsw_base_kernel(
    const float* __restrict__ C1, const float* __restrict__ D12,
    float* __restrict__ B) {
  int i = blockIdx.x * 256 + threadIdx.x;          // 384*128 elements
  int k = i >> 7, n = i & 127;
  float v = (k < NX) ? C1[n * NX + k] : D12[n * NU + (k - NX)];
  B[bsw_idx(k, n, 8)] = v;
}

__global__ __launch_bounds__(256) void fill_bsw_out_y_kernel(
    const float* __restrict__ C2, const float* __restrict__ D21,
    const float* __restrict__ D22, float* __restrict__ B) {
  int i = blockIdx.x * 256 + threadIdx.x;          // 512*128 elements
  int k = i >> 7, ny = i & 127;
  float v;
  if (k < 256)      v = C2 [ny * 256 + k];
  else if (k < 384) v = D21[ny * 128 + (k - 256)];
  else              v = D22[ny * 128 + (k - 384)];
  B[bsw_idx(k, 256 + ny, 24)] = v;
}

// ---------------- Newton-Schulz inverse: X <- X(2I - E X) ----------------
__global__ __launch_bounds__(256) void newton_T_kernel(
    const float* __restrict__ E, const float* __restrict__ X,
    float* __restrict__ T) {
  int c = threadIdx.x, r = blockIdx.x;
  float acc = 0.0f;
#pragma unroll 8
  for (int j = 0; j < 256; ++j) acc = fmaf(E[r * 256 + j], X[j * 256 + c], acc);
  T[r * 256 + c] = acc;
}

__global__ __launch_bounds__(256) void newton_update_kernel(
    const float* __restrict__ X, const float* __restrict__ T,
    float* __restrict__ Xn) {
  int c = threadIdx.x, r = blockIdx.x;
  float acc = 0.0f;
#pragma unroll 8
  for (int j = 0; j < 256; ++j) acc = fmaf(X[r * 256 + j], T[j * 256 + c], acc);
  Xn[r * 256 + c] = 2.0f * X[r * 256 + c] - acc;   // X(2I - T)
}

// dx columns of swizzled out-weights: Bsw(k,n) = (Einv @ [F|B1|B2])[n][k]
__global__ __launch_bounds__(256) void fill_bsw_out_dx_kernel(
    const float* __restrict__ X, const float* __restrict__ F,
    const float* __restrict__ B1, const float* __restrict__ B2,
    float* __restrict__ B) {
  int n = blockIdx.x;                               // 0..255
#pragma unroll 1
  for (int half = 0; half < 2; ++half) {
    int k = threadIdx.x + half * 256;               // 0..511, wave-uniform branches
    float acc = 0.0f;
    if (k < 256) {
#pragma unroll 8
      for (int j = 0; j < 256; ++j) acc = fmaf(X[n * 256 + j], F[j * 256 + k], acc);
    } else if (k < 384) {
      int kc = k - 256;
#pragma unroll 8
      for (int j = 0; j < 256; ++j) acc = fmaf(X[n * 256 + j], B1[j * 128 + kc], acc);
    } else {
      int kc = k - 384;
#pragma unroll 8
      for (int j = 0; j < 256; ++j) acc = fmaf(X[n * 256 + j], B2[j * 128 + kc], acc);
    }
    B[bsw_idx(k, n, 24)] = acc;
  }
}

// ---------------- WMMA f32 GEMM:  out[m][n] = sum_k A[m][k]*Bsw(k,n) --------
// Block: 256 threads = 8 waves (2 M x 4 N), M-block = 64 rows.
// A double-buffered in LDS via async global->LDS copies; K chunked by 64
// (every chunk comes from a single source matrix).
#define KC 64
#define APITCH 68            // (4*row + k) banks: half-waves on disjoint mod-4 groups
#define ABUF (64 * APITCH)   // 4352 floats = 17408 B per buffer

template <int NF, int NCHUNK, int STORE_MODE>
__global__ __launch_bounds__(256) void gemm_wmma_kernel(
    AChunks ac, const float* __restrict__ Bsw, int nftot,
    float* __restrict__ out0, float* __restrict__ out1) {
  __shared__ float As[2 * ABUF];   // 34816 B
  const int tid = threadIdx.x;
  const int lane = tid & 31;
  const int waveId = tid >> 5;
  const int wm = waveId & 1;        // M half (0/1 -> rows 0-31 / 32-63)
  const int wn = waveId >> 1;       // N quarter
  const size_t m0blk = (size_t)blockIdx.x * 64;

  v8f acc[2][NF];
#pragma unroll
  for (int mf = 0; mf < 2; ++mf)
#pragma unroll
    for (int nf = 0; nf < NF; ++nf) acc[mf][nf] = {};

  // stage chunk `ch` into buffer `buf` (async, 4 x b128 per thread)
  auto stage = [&](int ch, int buf) {
    const float* __restrict__ src = ac.p[ch];
    const int ld = ac.ld[ch];
    float* dst = &As[buf * ABUF];
#pragma unroll
    for (int it = 0; it < 4; ++it) {
      int gi = it * 256 + tid;
      int row = gi >> 4;            // 16 float4 per row (KC=64)
      int c4 = gi & 15;
      copy16_async(src + (m0blk + row) * ld + c4 * 4, dst + row * APITCH + c4 * 4);
    }
  };

  stage(0, 0);
  wait_async0();
  __syncthreads();

  const int kk = (lane >> 4) * 2;   // upper half-wave holds K=2,3
  const int arow = (lane & 15);

#pragma unroll 1
  for (int ch = 0; ch < NCHUNK; ++ch) {
    if (ch + 1 < NCHUNK) stage(ch + 1, (ch + 1) & 1);   // prefetch next chunk

    const float* Ab = &As[(ch & 1) * ABUF];
    const int ksBase = ch * (KC / 4);
#pragma unroll
    for (int ks = 0; ks < KC / 4; ++ks) {
      int k = ks * 4;
      v2f a[2];
#pragma unroll
      for (int mf = 0; mf < 2; ++mf)
        a[mf] = *(const v2f*)&Ab[(wm * 32 + mf * 16 + arow) * APITCH + k + kk];
      v2f b[NF];
#pragma unroll
      for (int nf = 0; nf < NF; ++nf) {
        size_t idx = ((size_t)(ksBase + ks) * nftot + (wn * NF + nf)) * 32 + lane;
        b[nf] = ((const v2f*)Bsw)[idx];
      }
#pragma unroll
      for (int mf = 0; mf < 2; ++mf)
#pragma unroll
        for (int nf = 0; nf < NF; ++nf)
          acc[mf][nf] = __builtin_amdgcn_wmma_f32_16x16x4_f32(
              false, a[mf], false, b[nf], (short)0, acc[mf][nf], false, false);
    }

    if (ch + 1 < NCHUNK) {
      wait_async0();     // my slice of the prefetch landed
      __syncthreads();   // everyone's slice landed; prev buffer free for reuse
    }
  }

  // store: C/D layout -> VGPR e: lanes0-15 M=e, lanes16-31 M=e+8; N = lane&15
  const int mhalf = (lane >> 4) * 8;
  const int nlow = lane & 15;
#pragma unroll
  for (int mf = 0; mf < 2; ++mf)
#pragma unroll
    for (int nf = 0; nf < NF; ++nf) {
      int n0 = (wn * NF + nf) * 16;
      int coln = n0 + nlow;
#pragma unroll
      for (int e = 0; e < 8; ++e) {
        size_t row = m0blk + wm * 32 + mf * 16 + mhalf + e;
        float val = acc[mf][nf][e];
        if (STORE_MODE == 0) {
          out0[row * 128 + coln] = val;                        // base -> w buffer
        } else {
          if (n0 < 256) out0[row * 256 + coln] = val;          // dx
          else          out1[row * 128 + (coln - 256)] = val;  // y
        }
      }
    }
}

// ---------------- sequential tanh recursion (in place: base -> w) ----------
__global__ __launch_bounds__(64) void wsolve_kernel(
    float* __restrict__ wbuf, const float* __restrict__ D11,
    const float* __restrict__ lam, const float* __restrict__ bv) {
  __shared__ float wl[128 * 65];      // [col][row], padded
  const int tid = threadIdx.x;        // 0..63 : one batch row each
  const size_t m0 = (size_t)blockIdx.x * 64;

  // load 64 rows x 128 cols transposed into LDS (coalesced float4 reads)
#pragma unroll 4
  for (int it = 0; it < 32; ++it) {
    int gi = it * 64 + tid;
    int row = gi >> 5;                // 32 float4 per row
    int c4 = gi & 31;
    float4 v = *(const float4*)&wbuf[(m0 + row) * 128 + c4 * 4];
    wl[(c4 * 4 + 0) * 65 + row] = v.x;
    wl[(c4 * 4 + 1) * 65 + row] = v.y;
    wl[(c4 * 4 + 2) * 65 + row] = v.z;
    wl[(c4 * 4 + 3) * 65 + row] = v.w;
  }
  __syncthreads();

  // k-loop: position k holds base until overwritten with w[k] (exact scan
  // semantics: w[:,j>=k]==0 at step k, so only j<k contribute).
  // 4 partial accumulators break the serial FMA latency chain.
  for (int k = 0; k < NQ; ++k) {
    const float* d11row = D11 + (size_t)k * NQ;    // wave-uniform -> s_loads
    float a0 = 0.0f, a1 = 0.0f, a2 = 0.0f, a3 = 0.0f;
    int j = 0;
    for (; j + 3 < k; j += 4) {
      a0 = fmaf(wl[(j + 0) * 65 + tid], d11row[j + 0], a0);
      a1 = fmaf(wl[(j + 1) * 65 + tid], d11row[j + 1], a1);
      a2 = fmaf(wl[(j + 2) * 65 + tid], d11row[j + 2], a2);
      a3 = fmaf(wl[(j + 3) * 65 + tid], d11row[j + 3], a3);
    }
    for (; j < k; ++j) a0 = fmaf(wl[j * 65 + tid], d11row[j], a0);
    float acc = wl[k * 65 + tid] + bv[k] + ((a0 + a1) + (a2 + a3));
    float wv = tanhf(acc / lam[k]);
    wl[k * 65 + tid] = wv;            // column `tid` is thread-exclusive
  }
  __syncthreads();

  // write back transposed (coalesced float4 stores)
#pragma unroll 4
  for (int it = 0; it < 32; ++it) {
    int gi = it * 64 + tid;
    int row = gi >> 5;
    int c4 = gi & 31;
    float4 v;
    v.x = wl[(c4 * 4 + 0) * 65 + row];
    v.y = wl[(c4 * 4 + 1) * 65 + row];
    v.z = wl[(c4 * 4 + 2) * 65 + row];
    v.w = wl[(c4 * 4 + 3) * 65 + row];
    *(float4*)&wbuf[(m0 + row) * 128 + c4 * 4] = v;
  }
}

// ---------------------------------------------------------------------------
extern "C" void kernel_launch(void* const* d_in, const int* in_sizes, int n_in,
                              void* d_out, int out_size, void* d_ws, size_t ws_size,
                              hipStream_t stream) {
  const float* x   = (const float*)d_in[0];
  const float* u   = (const float*)d_in[1];
  const float* F   = (const float*)d_in[2];
  const float* B1  = (const float*)d_in[3];
  const float* B2  = (const float*)d_in[4];
  const float* C1  = (const float*)d_in[5];
  const float* C2  = (const float*)d_in[6];
  const float* D11 = (const float*)d_in[7];
  const float* D12 = (const float*)d_in[8];
  const float* D21 = (const float*)d_in[9];
  const float* D22 = (const float*)d_in[10];
  const float* E   = (const float*)d_in[11];
  const float* Lam = (const float*)d_in[12];
  const float* bv  = (const float*)d_in[13];

  float* ws   = (float*)d_ws;
  float* wbuf = ws;                               // NB*NQ   (base, then w)
  float* Xp   = wbuf + (size_t)NBATCH * NQ;       // 256*256
  float* Xq   = Xp + NX * NX;
  float* Tm   = Xq + NX * NX;
  float* BswO = Tm + NX * NX;                     // 128*24*64 = 196608
  float* BswB = BswO + 196608;                    // 96*8*64   = 49152

  init_X_kernel<<<(NX * NX) / 256, 256, 0, stream>>>(Xp);
  fill_bsw_base_kernel<<<(384 * 128) / 256, 256, 0, stream>>>(C1, D12, BswB);
  fill_bsw_out_y_kernel<<<(512 * 128) / 256, 256, 0, stream>>>(C2, D21, D22, BswO);

  // Newton-Schulz: ||I - E|| ~ 0.2 => 5 iterations reach f32 precision
  float* Xa = Xp; float* Xb = Xq;
  for (int it = 0; it < 5; ++it) {
    newton_T_kernel<<<NX, NX, 0, stream>>>(E, Xa, Tm);
    newton_update_kernel<<<NX, NX, 0, stream>>>(Xa, Tm, Xb);
    float* t = Xa; Xa = Xb; Xb = t;
  }
  fill_bsw_out_dx_kernel<<<NX, 256, 0, stream>>>(Xa, F, B1, B2, BswO);

  // base = [x|u] @ [C1|D12]^T   (K = 384, N = 128), chunks of 64 columns
  AChunks acB;
  acB.p[0] = x;        acB.ld[0] = NX;
  acB.p[1] = x + 64;   acB.ld[1] = NX;
  acB.p[2] = x + 128;  acB.ld[2] = NX;
  acB.p[3] = x + 192;  acB.ld[3] = NX;
  acB.p[4] = u;        acB.ld[4] = NU;
  acB.p[5] = u + 64;   acB.ld[5] = NU;
  acB.p[6] = x;        acB.ld[6] = NX;   // unused
  acB.p[7] = x;        acB.ld[7] = NX;   // unused
  gemm_wmma_kernel<2, 6, 0><<<NBATCH / 64, 256, 0, stream>>>(acB, BswB, 8, wbuf, nullptr);

  wsolve_kernel<<<NBATCH / 64, 64, 0, stream>>>(wbuf, D11, Lam, bv);

  // [dx|y] = [x|w|u] @ Wout^T   (K = 512, N = 384), chunks of 64 columns
  AChunks acO;
  acO.p[0] = x;           acO.ld[0] = NX;
  acO.p[1] = x + 64;      acO.ld[1] = NX;
  acO.p[2] = x + 128;     acO.ld[2] = NX;
  acO.p[3] = x + 192;     acO.ld[3] = NX;
  acO.p[4] = wbuf;        acO.ld[4] = NQ;
  acO.p[5] = wbuf + 64;   acO.ld[5] = NQ;
  acO.p[6] = u;           acO.ld[6] = NU;
  acO.p[7] = u + 64;      acO.ld[7] = NU;
  float* dx = (float*)d_out;
  float* y  = dx + (size_t)NBATCH * NX;
  gemm_wmma_kernel<6, 8, 1><<<NBATCH / 64, 256, 0, stream>>>(acO, BswO, 24, dx, y);
}